// CountSketch_1769526526742
// MI455X (gfx1250) — compile-verified
//
#include <hip/hip_runtime.h>

#define D_IN   8192
#define D_FEAT 2048
#define BATCH  16384

#define ROWS_PER_BLOCK 2
#define CHUNK  1024                 // floats of x per row per stage (4 KB)
#define NCHUNK (D_IN / CHUNK)       // 8
#define THREADS 256                 // 8 waves (wave32)

__global__ __launch_bounds__(THREADS)
void countsketch_kernel(const float* __restrict__ x,
                        const float* __restrict__ s_hash,
                        const int*   __restrict__ i_hash,
                        float* __restrict__ out)
{
    // 16 KB accumulator + 16 KB double-buffered x staging = 32 KB / block
    __shared__ __align__(16) float acc[ROWS_PER_BLOCK][D_FEAT];
    __shared__ __align__(16) float xs[2][ROWS_PER_BLOCK][CHUNK];

    const unsigned t    = threadIdx.x;
    const unsigned row0 = blockIdx.x * ROWS_PER_BLOCK;
    const unsigned t4   = t * 4u;

    // Zero the LDS accumulator (ds_store_b128)
    float4* accv = reinterpret_cast<float4*>(&acc[0][0]);
#pragma unroll
    for (int i = 0; i < (ROWS_PER_BLOCK * D_FEAT / 4) / THREADS; ++i)
        accv[t + i * THREADS] = make_float4(0.f, 0.f, 0.f, 0.f);

    // Async HBM -> LDS copy of one chunk (CHUNK floats x ROWS_PER_BLOCK rows).
    // Each lane moves 16B; 256 lanes cover 4 KB = one row-chunk per call.
    // ASYNCcnt-tracked; per-wave in-order completion lets us double-buffer.
    auto issue = [&](unsigned c, unsigned buf) {
        unsigned lds0 = (unsigned)(unsigned long long)(&xs[buf][0][t4]);
        unsigned lds1 = (unsigned)(unsigned long long)(&xs[buf][1][t4]);
        unsigned g0   = (row0 * D_IN + c * CHUNK + t4) * 4u;   // byte offset < 2^31
        unsigned g1   = g0 + D_IN * 4u;
        asm volatile("global_load_async_to_lds_b128 %0, %1, %2"
                     :: "v"(lds0), "v"(g0), "s"(x) : "memory");
        asm volatile("global_load_async_to_lds_b128 %0, %1, %2"
                     :: "v"(lds1), "v"(g1), "s"(x) : "memory");
    };

    issue(0, 0);  // prologue: chunk 0 -> buffer 0

    for (unsigned c = 0; c < NCHUNK; ++c) {
        const unsigned cur = c & 1u;

        if (c + 1 < NCHUNK) {
            issue(c + 1, cur ^ 1u);                       // prefetch next chunk
            // 4 outstanding per wave after issue; <=2 left means chunk c landed
            asm volatile("s_wait_asynccnt 0x2" ::: "memory");
        } else {
            asm volatile("s_wait_asynccnt 0x0" ::: "memory");
        }
        __syncthreads();  // chunk c visible to all waves; acc zeroed (c==0)

        // ---- consume chunk c: one hash load amortized over both rows ----
        const float4 s4 = reinterpret_cast<const float4*>(s_hash)[c * (CHUNK / 4) + t];
        const int4   i4 = reinterpret_cast<const int4*>(i_hash)[c * (CHUNK / 4) + t];
        const float4 x0 = reinterpret_cast<const float4*>(&xs[cur][0][0])[t];
        const float4 x1 = reinterpret_cast<const float4*>(&xs[cur][1][0])[t];

        atomicAdd(&acc[0][i4.x], s4.x * x0.x);
        atomicAdd(&acc[0][i4.y], s4.y * x0.y);
        atomicAdd(&acc[0][i4.z], s4.z * x0.z);
        atomicAdd(&acc[0][i4.w], s4.w * x0.w);
        atomicAdd(&acc[1][i4.x], s4.x * x1.x);
        atomicAdd(&acc[1][i4.y], s4.y * x1.y);
        atomicAdd(&acc[1][i4.z], s4.z * x1.z);
        atomicAdd(&acc[1][i4.w], s4.w * x1.w);

        __syncthreads();  // protect buffer `cur` before it is refilled
    }

    // ---- coalesced 128-bit write-out of the accumulator ----
#pragma unroll
    for (int r = 0; r < ROWS_PER_BLOCK; ++r) {
        const float4* av = reinterpret_cast<const float4*>(&acc[r][0]);
        float4* ov = reinterpret_cast<float4*>(out + (size_t)(row0 + r) * D_FEAT);
#pragma unroll
        for (int i = 0; i < (D_FEAT / 4) / THREADS; ++i)
            ov[t + i * THREADS] = av[t + i * THREADS];
    }
}

extern "C" void kernel_launch(void* const* d_in, const int* in_sizes, int n_in,
                              void* d_out, int out_size, void* d_ws, size_t ws_size,
                              hipStream_t stream)
{
    (void)in_sizes; (void)n_in; (void)out_size; (void)d_ws; (void)ws_size;
    const float* x      = (const float*)d_in[0];
    const float* s_hash = (const float*)d_in[1];
    const int*   i_hash = (const int*)d_in[2];
    float* out = (float*)d_out;

    countsketch_kernel<<<dim3(BATCH / ROWS_PER_BLOCK), dim3(THREADS), 0, stream>>>(
        x, s_hash, i_hash, out);
}